// Kernel_14216341749914
// MI455X (gfx1250) — compile-verified
//
#include <hip/hip_runtime.h>

// MI455X / gfx1250 fused equivariant-kernel GEMM:
//   out[n, j] = sum_d ( sc[n, cmap(d)] * ang[n, amap(d)] ) * TP[j, d]
//   sc = (sk @ weights) / sqrt(12)
// Write-BW bound (346MB out @ 23.3TB/s -> ~16.5us floor). fp32 WMMA 16x16x4,
// double-buffered async-LDS TP staging, 2-tile N-blocking per wave (A reuse).

typedef float v2f __attribute__((ext_vector_type(2)));
typedef float v8f __attribute__((ext_vector_type(8)));

#define NROWS   32000
#define FDIM    2704
#define NT      169         // 16-col output tiles (2704/16)
#define FC      720
#define ADIM    25
#define SB      12
#define KC      52          // K-chunk (2704 = 52 * 52)
#define NCHUNK  52
#define TPSTR   56          // padded row stride (floats) for 52-wide tiles
#define SCSTR   721         // padded sc row stride (gcd(721,64)=1 -> conflict-free)
#define ANGSTR  33

// LDS layout (float offsets)
#define OFF_SC    0                         // 16 x 721          = 11536
#define OFF_ANG   11536                     // 16 x 33           =   528
#define OFF_MAP   12064                     // 2704 packed u32   =  2704
#define OFF_SK    14768                     // 16 x 12           =   192
#define OFF_FILT  14960                     // 16 x 56           =   896
#define OFF_TP0   15856                     // 256 x 56          = 14336
#define OFF_TP1   30192                     // 256 x 56          = 14336
#define OFF_W     15856                     // 12 x 720 overlaid on TP bufs (phase 1 only)
#define LDS_FLOATS 44528                    // ~174 KB (<= 320 KB/WGP)

// ragged rep structure: (l, C_l, m=2l+1) -> d-segments
__device__ __forceinline__ unsigned pack_map(int d) {
    int c, a;
    if (d < 144)       { c = d;                    a = 0; }
    else if (d < 960)  { int t = d - 144;  c = 144 + t / 3; a = 1  + t % 3; }
    else if (d < 2000) { int t = d - 960;  c = 416 + t / 5; a = 4  + t % 5; }
    else if (d < 2560) { int t = d - 2000; c = 624 + t / 7; a = 9  + t % 7; }
    else               { int t = d - 2560; c = 704 + t / 9; a = 16 + t % 9; }
    return ((unsigned)c << 5) | (unsigned)a;
}

// generic LDS pointer -> 32-bit LDS byte address (flat LDS aperture keeps the
// byte offset in addr[31:0] per CDNA5 ISA flat-addressing rules)
__device__ __forceinline__ unsigned lds_addr(const void* p) {
    return (unsigned)(unsigned long long)p;
}

// stage one 256x52 TP tile into LDS with async DMA.
// 256 rows * 13 b128 segs = 3328 = 256 threads * 13 -> exactly 13 async ops per
// thread/wave, so the s_wait_asynccnt immediate below is exact.
__device__ __forceinline__ void issue_tp_chunk(const float* __restrict__ tp,
                                               int j0base, int kc,
                                               unsigned lds_base, int tid) {
#pragma unroll
    for (int t = 0; t < 13; ++t) {
        int idx = tid + t * 256;
        int row = idx / 13;
        int seg = idx - row * 13;
        int grow = j0base + row;
        if (grow > FDIM - 1) grow = FDIM - 1;  // clamp tail block rows
        unsigned voff  = (unsigned)((grow * FDIM + kc) * 4 + seg * 16);
        unsigned laddr = lds_base + (unsigned)(row * TPSTR * 4 + seg * 16);
        asm volatile("global_load_async_to_lds_b128 %0, %1, %2"
                     :: "v"(laddr), "v"(voff), "s"(tp) : "memory");
    }
}

__global__ __launch_bounds__(256)
void fused_equivariant_gemm(const float* __restrict__ sk_g,
                            const float* __restrict__ ang_g,
                            const float* __restrict__ w_g,
                            const float* __restrict__ tp_g,
                            float* __restrict__ out_g) {
    extern __shared__ float smem[];
    unsigned* mapl = (unsigned*)(smem + OFF_MAP);

    const int tid    = threadIdx.x;
    const int bx     = blockIdx.x;          // col super-tile (256 cols)
    const int by     = blockIdx.y;          // row tile (16 rows)
    const int n0     = by * 16;
    const int j0base = bx * 256;

    // ---- phase 0: stage sk, ang, scaled weights, (c,a) map ----
    if (tid < 16 * SB) smem[OFF_SK + tid] = sk_g[n0 * SB + tid];
    for (int idx = tid; idx < 16 * ADIM; idx += 256) {
        int r = idx / ADIM, a = idx - r * ADIM;
        smem[OFF_ANG + r * ANGSTR + a] = ang_g[(n0 + r) * ADIM + a];
    }
    for (int d = tid; d < FDIM; d += 256) mapl[d] = pack_map(d);
    const float wscale = 0.28867513459481287f;   // 1/sqrt(12)
    for (int idx = tid; idx < SB * FC; idx += 256)
        smem[OFF_W + idx] = w_g[idx] * wscale;
    __syncthreads();

    // ---- phase 1: sc[16 x 720] = sk @ W' (stays in LDS forever) ----
    for (int idx = tid; idx < 16 * FC; idx += 256) {
        int r = idx / FC, c = idx - r * FC;
        float s = 0.f;
#pragma unroll
        for (int k = 0; k < SB; ++k)
            s += smem[OFF_SK + r * SB + k] * smem[OFF_W + k * FC + c];
        smem[OFF_SC + r * SCSTR + c] = s;
    }
    __syncthreads();   // weights region dead -> becomes TP double buffer

    const unsigned ldsTp0 = lds_addr(smem + OFF_TP0);
    const unsigned ldsTp1 = lds_addr(smem + OFF_TP1);
    issue_tp_chunk(tp_g, j0base, 0, ldsTp0, tid);   // prologue DMA

    const int wave = tid >> 5;
    const int lane = tid & 31;
    const int r    = lane & 15;                 // A row / B col within tile
    const int khi  = (lane >> 4) << 1;          // lanes 16-31 hold K+2/K+3
    const int jt0  = bx * 16 + wave;            // wave's first 16-col tile
    const int jt1  = jt0 + 8;                   // wave's second 16-col tile
    const int brow0 = (wave * 16 + r) * TPSTR;          // B LDS row, tile 0
    const int brow1 = brow0 + 128 * TPSTR;              // B LDS row, tile 1
    v8f acc0 = {}, acc1 = {};

    // ---- phase 2: K loop, double-buffered async TP + on-the-fly filt ----
    for (int ch = 0; ch < NCHUNK; ++ch) {
        const int kc = ch * KC;

        // build this chunk's filt[16 x 52] panel in registers (stable sources)
        float fv[4]; int fok[4], fidx[4];
#pragma unroll
        for (int t = 0; t < 4; ++t) {
            int idx = tid + t * 256;            // idx = rr*64 + kk
            int kk  = idx & 63;
            int rr  = idx >> 6;
            int ok  = kk < KC;
            int kkc = ok ? kk : 0;
            unsigned p = mapl[kc + kkc];
            fv[t]   = smem[OFF_SC + rr * SCSTR + (int)(p >> 5)] *
                      smem[OFF_ANG + rr * ANGSTR + (int)(p & 31u)];
            fok[t]  = ok;
            fidx[t] = rr * TPSTR + kk;
        }

        __syncthreads();                        // prev iter done reading filt/tp
#pragma unroll
        for (int t = 0; t < 4; ++t)
            if (fok[t]) smem[OFF_FILT + fidx[t]] = fv[t];

        if (ch + 1 < NCHUNK) {
            issue_tp_chunk(tp_g, j0base, kc + KC,
                           ((ch + 1) & 1) ? ldsTp1 : ldsTp0, tid);
            asm volatile("s_wait_asynccnt 13" ::: "memory"); // chunk ch landed
        } else {
            asm volatile("s_wait_asynccnt 0" ::: "memory");
        }
        __syncthreads();                        // filt stored + all waves' DMA in

        const float* fb = smem + OFF_FILT;
        const float* tb = smem + ((ch & 1) ? OFF_TP1 : OFF_TP0);
        if (jt1 < NT) {                         // both tiles live (uniform/wave)
#pragma unroll
            for (int kk = 0; kk < KC; kk += 4) {
                // A 16x4 f32: VGPR0 = K+0 (lanes 0-15) / K+2 (lanes 16-31)
                v2f a  = *(const v2f*)(fb + r * TPSTR + kk + khi);
                v2f b0 = *(const v2f*)(tb + brow0 + kk + khi);
                v2f b1 = *(const v2f*)(tb + brow1 + kk + khi);
                acc0 = __builtin_amdgcn_wmma_f32_16x16x4_f32(
                           false, a, false, b0, (short)0, acc0, false, false);
                acc1 = __builtin_amdgcn_wmma_f32_16x16x4_f32(
                           false, a, false, b1, (short)0, acc1, false, false);
            }
        } else if (jt0 < NT) {                  // tail super-tile: tile 0 only
#pragma unroll
            for (int kk = 0; kk < KC; kk += 4) {
                v2f a  = *(const v2f*)(fb + r * TPSTR + kk + khi);
                v2f b0 = *(const v2f*)(tb + brow0 + kk + khi);
                acc0 = __builtin_amdgcn_wmma_f32_16x16x4_f32(
                           false, a, false, b0, (short)0, acc0, false, false);
            }
        }
    }

    // ---- epilogue: C/D layout -> lanes 0-15 rows 0-7, lanes 16-31 rows 8-15
    const int rbase = n0 + ((lane >> 4) << 3);
    if (jt0 < NT) {
        const int jc0 = jt0 * 16 + (lane & 15);
#pragma unroll
        for (int v = 0; v < 8; ++v)
            out_g[(size_t)(rbase + v) * FDIM + jc0] = acc0[v];
    }
    if (jt1 < NT) {
        const int jc1 = jt1 * 16 + (lane & 15);
#pragma unroll
        for (int v = 0; v < 8; ++v)
            out_g[(size_t)(rbase + v) * FDIM + jc1] = acc1[v];
    }
}

extern "C" void kernel_launch(void* const* d_in, const int* in_sizes, int n_in,
                              void* d_out, int out_size, void* d_ws, size_t ws_size,
                              hipStream_t stream) {
    const float* sk  = (const float*)d_in[0];   // (16,16,125,12)  -> [32000][12]
    const float* ang = (const float*)d_in[1];   // (32000, 25)
    const float* w   = (const float*)d_in[2];   // (12, 720)
    const float* tp  = (const float*)d_in[3];   // (2704, 2704)
    float* out = (float*)d_out;                 // (16,16,125,52,52) -> [32000][2704]

    dim3 grid(11, NROWS / 16);                  // 11 col super-tiles x 2000 row tiles
    size_t shmem = (size_t)LDS_FLOATS * sizeof(float);
    fused_equivariant_gemm<<<grid, 256, shmem, stream>>>(sk, ang, w, tp, out);
}